// Model_29592324669787
// MI455X (gfx1250) — compile-verified
//
#include <hip/hip_runtime.h>
#include <hip/hip_bf16.h>
#include <math.h>

#define IN_DIM   512
#define OUT_DIM  128
#define LDA      516   // 512 + 4 pad: LDS bank-conflict-free A reads (bank = row*4 + col)

typedef __attribute__((ext_vector_type(2))) float v2f;
typedef __attribute__((ext_vector_type(8))) float v8f;

// ---------------------------------------------------------------------------
// GEMM: XW[n,128] = X[n,512] @ W[512,128], via V_WMMA_F32_16X16X4_F32.
// Block = 256 thr (8 waves), tile = 32 rows x 128 cols.
// Wave w owns col tile [16w,16w+16) for BOTH 16-row halves, sharing one B
// fragment across two WMMAs (halves W load traffic per WMMA).
// ---------------------------------------------------------------------------
__global__ __launch_bounds__(256) void gemm_f32_wmma(const float* __restrict__ X,
                                                     const float* __restrict__ W,
                                                     float* __restrict__ XW,
                                                     int nRows) {
    __shared__ float As[32 * LDA];   // 66 KB (gfx1250: up to 320 KB/WG)

    const int tid     = threadIdx.x;
    const int rowbase = blockIdx.x * 32;

    // Stage A tile: 32x512 f32 = 4096 float4, 16 per thread.
    for (int i = tid; i < 32 * 128; i += 256) {
        const int r  = i >> 7;      // row 0..31
        const int c4 = i & 127;     // float4 col
        float4 v = make_float4(0.f, 0.f, 0.f, 0.f);
        const int gr = rowbase + r;
        if (gr < nRows)
            v = *(const float4*)(X + (size_t)gr * IN_DIM + c4 * 4);
        *(float4*)(&As[r * LDA + c4 * 4]) = v;
    }
    __syncthreads();

    const int wave = tid >> 5;
    const int lane = tid & 31;
    const int half = lane >> 4;     // 0: K-pair {0,1}, 1: K-pair {2,3}
    const int l16  = lane & 15;
    const int colbase = wave * 16;

    v8f acc0 = {0.f, 0.f, 0.f, 0.f, 0.f, 0.f, 0.f, 0.f};
    v8f acc1 = {0.f, 0.f, 0.f, 0.f, 0.f, 0.f, 0.f, 0.f};

    // A fragments: lane -> A[M = l16 (+16), k + half*2 + {0,1}]
    const float* Arow0 = &As[l16 * LDA + half * 2];
    const float* Arow1 = &As[(l16 + 16) * LDA + half * 2];
    // B fragment: lane -> W[k + half*2 + {0,1}, colbase + l16]
    const float* Wcol = W + (size_t)(half * 2) * OUT_DIM + colbase + l16;

#pragma unroll 4
    for (int k = 0; k < IN_DIM; k += 4) {
        const v2f a0 = *(const v2f*)(Arow0 + k);   // 8B-aligned (even float index)
        const v2f a1 = *(const v2f*)(Arow1 + k);
        v2f b;
        b[0] = Wcol[(size_t)k * OUT_DIM];
        b[1] = Wcol[(size_t)k * OUT_DIM + OUT_DIM];
        acc0 = __builtin_amdgcn_wmma_f32_16x16x4_f32(
            false, a0, false, b, (short)0, acc0, false, false);
        acc1 = __builtin_amdgcn_wmma_f32_16x16x4_f32(
            false, a1, false, b, (short)0, acc1, false, false);
    }

    // C layout: VGPR r -> row = tilebase + r + half*8, col = colbase + l16
    float* out0 = XW + (size_t)(rowbase + half * 8) * OUT_DIM + colbase + l16;
    float* out1 = XW + (size_t)(rowbase + 16 + half * 8) * OUT_DIM + colbase + l16;
#pragma unroll
    for (int r = 0; r < 8; ++r) {
        out0[(size_t)r * OUT_DIM] = acc0[r];
        out1[(size_t)r * OUT_DIM] = acc1[r];
    }
}

// ---------------------------------------------------------------------------
// Edge scatter-add: agg[dst[e], :] += xw[src[e], :].  One wave per edge.
// ---------------------------------------------------------------------------
__global__ __launch_bounds__(256) void edge_scatter(const float* __restrict__ xw,
                                                    const int* __restrict__ src,
                                                    const int* __restrict__ dst,
                                                    float* __restrict__ agg,
                                                    int nEdges) {
    const int e = blockIdx.x * 8 + (threadIdx.x >> 5);
    if (e >= nEdges) return;
    const int lane = threadIdx.x & 31;
    const int s = __builtin_amdgcn_readfirstlane(src[e]);   // wave-uniform -> SGPR
    const int d = __builtin_amdgcn_readfirstlane(dst[e]);
    const float4 v = *(const float4*)(xw + (size_t)s * OUT_DIM + lane * 4);
    float* p = agg + (size_t)d * OUT_DIM + lane * 4;
    unsafeAtomicAdd(p + 0, v.x);   // global_atomic_add_f32, no return
    unsafeAtomicAdd(p + 1, v.y);
    unsafeAtomicAdd(p + 2, v.z);
    unsafeAtomicAdd(p + 3, v.w);
}

// ---------------------------------------------------------------------------
// Column sums over nodes (for mean): sum[j] += sum_n h[n, j]
// ---------------------------------------------------------------------------
__global__ __launch_bounds__(128) void colsum(const float* __restrict__ h,
                                              float* __restrict__ sum, int n) {
    const int j = threadIdx.x;
    float acc = 0.f;
    for (int node = blockIdx.x; node < n; node += gridDim.x)
        acc += h[(size_t)node * OUT_DIM + j];
    unsafeAtomicAdd(&sum[j], acc);
}

// ---------------------------------------------------------------------------
// Small prep (1 block, 128 thr): c = sigmoid(sum/N + b); u = Wb @ c;
// kc[0] = b2·u1 + bb, kc[1] = b1·u2 + bb.
// ---------------------------------------------------------------------------
__global__ __launch_bounds__(128) void prep(const float* __restrict__ sum1,
                                            const float* __restrict__ sum2,
                                            const float* __restrict__ b1,
                                            const float* __restrict__ b2,
                                            const float* __restrict__ Wb,
                                            const float* __restrict__ bb,
                                            float* __restrict__ u1,
                                            float* __restrict__ u2,
                                            float* __restrict__ kc, int n) {
    __shared__ float c1s[128], c2s[128], r1s[128], r2s[128];
    const int j = threadIdx.x;
    const float invn = 1.f / (float)n;
    c1s[j] = 1.f / (1.f + expf(-(sum1[j] * invn + b1[j])));
    c2s[j] = 1.f / (1.f + expf(-(sum2[j] * invn + b2[j])));
    __syncthreads();
    float a1 = 0.f, a2 = 0.f;
#pragma unroll 8
    for (int e = 0; e < 128; ++e) {
        const float w = Wb[j * 128 + e];
        a1 += w * c1s[e];
        a2 += w * c2s[e];
    }
    u1[j] = a1;
    u2[j] = a2;
    r1s[j] = b2[j] * a1;   // toward k1 = b2·u1 + bb
    r2s[j] = b1[j] * a2;   // toward k2 = b1·u2 + bb
    __syncthreads();
    if (j == 0) {
        float s1 = 0.f, s2 = 0.f;
        for (int e = 0; e < 128; ++e) { s1 += r1s[e]; s2 += r2s[e]; }
        kc[0] = s1 + bb[0];
        kc[1] = s2 + bb[0];
    }
}

// ---------------------------------------------------------------------------
// Score: out[n] = agg[n,:]·u + kc[kidx].  One wave per node.
// ---------------------------------------------------------------------------
__global__ __launch_bounds__(256) void score(const float* __restrict__ h,
                                             const float* __restrict__ u,
                                             const float* __restrict__ kc, int kidx,
                                             float* __restrict__ out, int n) {
    const int node = blockIdx.x * 8 + (threadIdx.x >> 5);
    if (node >= n) return;
    const int lane = threadIdx.x & 31;
    const float4 hv = *(const float4*)(h + (size_t)node * OUT_DIM + lane * 4);
    const float4 uv = *(const float4*)(u + lane * 4);
    float p = hv.x * uv.x + hv.y * uv.y + hv.z * uv.z + hv.w * uv.w;
#pragma unroll
    for (int off = 16; off > 0; off >>= 1)
        p += __shfl_xor(p, off, 32);
    if (lane == 0) out[node] = p + kc[kidx];
}

__global__ void fill_zero(float* __restrict__ p, size_t count) {
    size_t i = (size_t)blockIdx.x * blockDim.x + threadIdx.x;
    const size_t stride = (size_t)gridDim.x * blockDim.x;
    for (; i < count; i += stride) p[i] = 0.f;
}

// ---------------------------------------------------------------------------
extern "C" void kernel_launch(void* const* d_in, const int* in_sizes, int n_in,
                              void* d_out, int out_size, void* d_ws, size_t ws_size,
                              hipStream_t stream) {
    const float* feat = (const float*)d_in[0];
    const float* shuf = (const float*)d_in[1];
    const int*   src1 = (const int*)d_in[2];
    const int*   dst1 = (const int*)d_in[3];
    const int*   src2 = (const int*)d_in[4];
    const int*   dst2 = (const int*)d_in[5];
    const float* W1   = (const float*)d_in[6];
    const float* b1   = (const float*)d_in[7];
    const float* W2   = (const float*)d_in[8];
    const float* b2   = (const float*)d_in[9];
    const float* Wb   = (const float*)d_in[10];
    const float* bb   = (const float*)d_in[11];
    float* out = (float*)d_out;

    const int n      = in_sizes[0] / IN_DIM;   // 100000
    const int nEdges = in_sizes[2];            // 1600000

    float* ws = (float*)d_ws;
    const size_t NF = (size_t)n * OUT_DIM;
    float* xw   = ws;                 // 51.2 MB, reused per conv
    float* aggA = ws + NF;            // h1 then h3
    float* aggB = ws + 2 * NF;        // h2 then h4
    float* sum1 = ws + 3 * NF;
    float* sum2 = sum1 + 128;
    float* u1   = sum2 + 128;
    float* u2   = u1 + 128;
    float* kc   = u2 + 128;           // 2 floats

    const int gemmGrid  = (n + 31) / 32;
    const int edgeGrid  = (nEdges + 7) / 8;
    const int scoreGrid = (n + 7) / 8;

    // Zero aggA, aggB, sums/u/kc in one pass (contiguous).
    fill_zero<<<4096, 256, 0, stream>>>(aggA, 2 * NF + 514);

    // h1 = GC(feat, W1, src1->dst1) ; h2 = GC(feat, W2, src2->dst2)
    gemm_f32_wmma<<<gemmGrid, 256, 0, stream>>>(feat, W1, xw, n);
    edge_scatter<<<edgeGrid, 256, 0, stream>>>(xw, src1, dst1, aggA, nEdges);
    gemm_f32_wmma<<<gemmGrid, 256, 0, stream>>>(feat, W2, xw, n);
    edge_scatter<<<edgeGrid, 256, 0, stream>>>(xw, src2, dst2, aggB, nEdges);

    // c1/c2 -> u1/u2 and bias constants
    colsum<<<1024, 128, 0, stream>>>(aggA, sum1, n);
    colsum<<<1024, 128, 0, stream>>>(aggB, sum2, n);
    prep<<<1, 128, 0, stream>>>(sum1, sum2, b1, b2, Wb, bb, u1, u2, kc, n);

    // sc1 = h2·u1 + k1 ; sc2 = h1·u2 + k2
    score<<<scoreGrid, 256, 0, stream>>>(aggB, u1, kc, 0, out + 0 * (size_t)n, n);
    score<<<scoreGrid, 256, 0, stream>>>(aggA, u2, kc, 1, out + 1 * (size_t)n, n);

    // h3 = GC(shuf, W1, src1->dst1) ; h4 = GC(shuf, W2, src2->dst2)
    fill_zero<<<4096, 256, 0, stream>>>(aggA, 2 * NF);
    gemm_f32_wmma<<<gemmGrid, 256, 0, stream>>>(shuf, W1, xw, n);
    edge_scatter<<<edgeGrid, 256, 0, stream>>>(xw, src1, dst1, aggA, nEdges);
    gemm_f32_wmma<<<gemmGrid, 256, 0, stream>>>(shuf, W2, xw, n);
    edge_scatter<<<edgeGrid, 256, 0, stream>>>(xw, src2, dst2, aggB, nEdges);

    // sc3 = h4·u1 + k1 ; sc4 = h3·u2 + k2
    score<<<scoreGrid, 256, 0, stream>>>(aggB, u1, kc, 0, out + 2 * (size_t)n, n);
    score<<<scoreGrid, 256, 0, stream>>>(aggA, u2, kc, 1, out + 3 * (size_t)n, n);
}